// Space2Features_38139309589166
// MI455X (gfx1250) — compile-verified
//
#include <hip/hip_runtime.h>

typedef float v4f __attribute__((ext_vector_type(4)));

#define B_  4
#define H_  512
#define W_  2048
#define WC_ 16

// out[b,i,h,j] = x[b,0,h,j+i] * z[b,0,h,j], 0 when j+i >= W.
// One 256-thread (8-wave) block per (b,h) row.
// x row staged in LDS via CDNA5 async global->LDS b128 copies (ASYNCcnt),
// zero-padded 16 floats past W so the boundary masking is free.
// Output (256 MB, > L2) streamed with non-temporal b128 stores.
__global__ __launch_bounds__(256)
void Space2Features_38139309589166_kernel(const float* __restrict__ x,
                                          const float* __restrict__ z,
                                          float* __restrict__ out)
{
    __shared__ __align__(16) float tile[W_ + 16];   // 8256 B per block

    const int tid = threadIdx.x;
    const int h   = blockIdx.x;
    const int b   = blockIdx.y;

    const long  row   = (long)b * H_ + h;
    const float* xrow = x + row * W_;
    const float* zrow = z + row * W_;
    float*       obase = out + ((long)b * WC_ * H_ + h) * (long)W_;

    // ---- async copy x row (2048 f32 = 512 x b128, 2 per lane) into LDS ----
    {
        // Low 32 bits of a flat LDS pointer are the LDS byte address.
        unsigned lds0 = (unsigned)(unsigned long long)(&tile[0]) + (unsigned)tid * 16u;
        const float* g0 = xrow + tid * 4;
        asm volatile("global_load_async_to_lds_b128 %0, %1, off"
                     :: "v"(lds0), "v"(g0) : "memory");
        unsigned lds1 = lds0 + 256u * 16u;
        const float* g1 = g0 + 256 * 4;
        asm volatile("global_load_async_to_lds_b128 %0, %1, off"
                     :: "v"(lds1), "v"(g1) : "memory");
    }
    if (tid < 16) tile[W_ + tid] = 0.0f;            // zero pad: kills j+i >= W
    asm volatile("s_wait_asynccnt 0" ::: "memory"); // my async data landed in LDS
    __syncthreads();                                // everyone's data visible

    // ---- compute: each thread owns chunks j0 = 4*tid and 4*(tid+256) ----
    #pragma unroll
    for (int c = 0; c < 2; ++c) {
        const int j0 = (tid + c * 256) * 4;

        const v4f z4 = *(const v4f*)(zrow + j0);    // 16B-aligned global load

        // 20-float sliding window from LDS: 5 aligned ds_load_b128
        float w[20];
        #pragma unroll
        for (int k = 0; k < 5; ++k) {
            v4f t = *(const v4f*)(&tile[j0 + k * 4]);
            w[k * 4 + 0] = t[0];
            w[k * 4 + 1] = t[1];
            w[k * 4 + 2] = t[2];
            w[k * 4 + 3] = t[3];
        }

        // all 16 shifts from registers; NT stores (output never re-read)
        #pragma unroll
        for (int i = 0; i < WC_; ++i) {
            v4f o;
            o[0] = w[i + 0] * z4[0];
            o[1] = w[i + 1] * z4[1];
            o[2] = w[i + 2] * z4[2];
            o[3] = w[i + 3] * z4[3];
            __builtin_nontemporal_store(o, (v4f*)(obase + (long)i * (H_ * W_) + j0));
        }
    }
}

extern "C" void kernel_launch(void* const* d_in, const int* in_sizes, int n_in,
                              void* d_out, int out_size, void* d_ws, size_t ws_size,
                              hipStream_t stream) {
    const float* x = (const float*)d_in[0];   // [4,1,512,2048] f32
    const float* z = (const float*)d_in[1];   // [4,1,512,2048] f32
    // d_in[2] = w_ceil (==16), compiled in.
    float* out = (float*)d_out;               // [4,16,512,2048] f32

    (void)in_sizes; (void)n_in; (void)out_size; (void)d_ws; (void)ws_size;

    dim3 grid(H_, B_);   // one block per (h, b) row
    dim3 block(256);     // 8 waves (wave32)
    Space2Features_38139309589166_kernel<<<grid, block, 0, stream>>>(x, z, out);
}